// MultiScaleGraphPropagate_71055938945741
// MI455X (gfx1250) — compile-verified
//
#include <hip/hip_runtime.h>

// ---- problem constants (from reference) ----
#define B_ 4
#define T_ 8
#define N_ 10000
#define F_ 32
#define L_ 3
#define E_ 160000
#define S_ (B_*T_*N_*F_)          // 10,240,000 floats per (k) slice

typedef float v2f __attribute__((ext_vector_type(2)));
typedef float v8f __attribute__((ext_vector_type(8)));

// out[0] = x ; out[1] = 0 ; out[2] = 0   (float4-vectorized, exact grid)
__global__ __launch_bounds__(256) void msgp_init(const float4* __restrict__ x,
                                                 float4* __restrict__ out) {
    int i = blockIdx.x * 256 + threadIdx.x;      // i in [0, S_/4)
    float4 v = x[i];
    float4 z = make_float4(0.f, 0.f, 0.f, 0.f);
    out[i]            = v;
    out[S_/4 + i]     = z;
    out[2*(S_/4) + i] = z;
}

// One wave handles a tile of 16 edges for one batch b, all T=8 timesteps, all 32 features.
// C/D-layout mapping: lane l, acc slot g  ->  edge (g + 8*(l>=16)) of the tile, feature col (l&15).
// Per-edge weight is broadcast into exactly that layout by a single
// V_WMMA_F32_16X16X4_F32:  D = A(16x4, w in column K=0) x B(4x16, all ones)  =>  D[m,n] = w[m].
__global__ __launch_bounds__(256) void msgp_hop(const float* __restrict__ xin,
                                                float*       __restrict__ xout,
                                                const float* __restrict__ ew,
                                                const int*   __restrict__ eidx) {
    const int lane = threadIdx.x & 31;
    const int wave = blockIdx.x * 8 + (threadIdx.x >> 5);   // 8 waves / 256-thread block
    const int tilesPerB = E_ / 16;                          // 10000, exact
    const int b  = wave / tilesPerB;
    const int e0 = (wave - b * tilesPerB) * 16;
    const int hi  = lane >> 4;                              // 0: rows 0-7, 1: rows 8-15
    const int col = lane & 15;                              // feature column (and +16 for 2nd half)

    // --- A matrix (16x4 f32): lanes 0-15 hold A[m,K0],A[m,K1]; lanes 16-31 hold K2,K3.
    // Put mean-over-L weight of edge (e0+m) into A[m,0]; everything else zero.
    v2f a; a.x = 0.f; a.y = 0.f;
    if (hi == 0) {
        const int e = e0 + lane;
        const float w3 = ew[(b*L_ + 0)*E_ + e]
                       + ew[(b*L_ + 1)*E_ + e]
                       + ew[(b*L_ + 2)*E_ + e];
        a.x = w3 * (1.0f / 3.0f);                           // combine='mean' over lags
    }
    // --- B matrix: every slot 1.0f -> logical all-ones regardless of layout.
    v2f bones; bones.x = 1.0f; bones.y = 1.0f;
    v8f c = {};
    // d[g] = w[e0 + g + 8*hi], broadcast across the 16 feature columns (documented C/D layout)
    v8f d = __builtin_amdgcn_wmma_f32_16x16x4_f32(false, a, false, bones,
                                                  (short)0, c, false, false);

    // Per-lane 32-bit BYTE offsets within one (b,t) slice, computed once per tile.
    unsigned soff[8], toff[8];
    #pragma unroll
    for (int g = 0; g < 8; ++g) {
        const int e = e0 + g + 8*hi;
        const unsigned tg = (unsigned)eidx[e];        // edge_index[0] = tgt
        const unsigned sr = (unsigned)eidx[E_ + e];   // edge_index[1] = src
        soff[g] = sr * (unsigned)(F_ * 4) + (unsigned)(col * 4);
        toff[g] = tg * (unsigned)(F_ * 4) + (unsigned)(col * 4);
    }

    for (int t = 0; t < T_; ++t) {
        const char* xt = (const char*)(xin  + (size_t)(b*T_ + t) * (size_t)(N_*F_));
        char*       ot = (char*)      (xout + (size_t)(b*T_ + t) * (size_t)(N_*F_));

        // Phase 1: issue all 16 gathers of this timestep as one clause
        // (max per-wave MLP; no full loadcnt drain between edges).
        float v0[8], v1[8];
        #pragma unroll
        for (int g = 0; g < 8; ++g) {
            v0[g] = *(const float*)(xt + soff[g]);
            v1[g] = *(const float*)(xt + soff[g] + 64);
        }
        // Phase 2: scale by the WMMA-broadcast weight, scatter with hw f32 atomics.
        #pragma unroll
        for (int g = 0; g < 8; ++g) {
            const float w = d[g];
            atomicAdd((float*)(ot + toff[g]),      w * v0[g]);
            atomicAdd((float*)(ot + toff[g] + 64), w * v1[g]);
        }
    }
}

extern "C" void kernel_launch(void* const* d_in, const int* in_sizes, int n_in,
                              void* d_out, int out_size, void* d_ws, size_t ws_size,
                              hipStream_t stream) {
    const float* x    = (const float*)d_in[0];   // (B,T,N,F) f32
    const float* ew   = (const float*)d_in[1];   // (B,L,E)   f32
    const int*   eidx = (const int*)  d_in[2];   // (2,E)     int32
    float* out  = (float*)d_out;                 // (KS,B,T,N,F) f32
    float* out1 = out + (size_t)S_;
    float* out2 = out + (size_t)2*S_;

    // out[0]=x, out[1]=out[2]=0
    msgp_init<<<(S_/4)/256, 256, 0, stream>>>((const float4*)x, (float4*)out);

    // 40000 edge-tiles per hop, 8 waves per block -> 5000 blocks (exact)
    const int blocks = (B_ * (E_/16)) / 8;
    msgp_hop<<<blocks, 256, 0, stream>>>(x,    out1, ew, eidx);   // hop 1
    msgp_hop<<<blocks, 256, 0, stream>>>(out1, out2, ew, eidx);   // hop 2
}